// HGTLayer_24489903522221
// MI455X (gfx1250) — compile-verified
//
#include <hip/hip_runtime.h>
#include <math.h>

#define D 128
#define H 8
#define DK 16
#define MSTRIPS 8            // 8 x 16 = 128 rows per block

typedef __attribute__((ext_vector_type(2))) float v2f;
typedef __attribute__((ext_vector_type(8))) float v8f;
typedef unsigned int u32x4 __attribute__((ext_vector_type(4)));
typedef int i32x4v __attribute__((ext_vector_type(4)));
typedef int i32x8v __attribute__((ext_vector_type(8)));

// ---------------------------------------------------------------- utilities
__global__ void fill_kernel(float* __restrict__ p, float v, long long n) {
  long long i = (long long)blockIdx.x * blockDim.x + threadIdx.x;
  if (i < n) p[i] = v;
}

__device__ inline void atomicMaxF(float* addr, float val) {
  // monotone float-max via int max (val>=0) / uint min (val<0); init must be -inf
  if (!(val < 0.0f)) atomicMax((int*)addr, __float_as_int(val));
  else               atomicMin((unsigned int*)addr, __float_as_uint(val));
}

__device__ inline void load16(const float* __restrict__ p, float* r) {
  const float4* q = (const float4*)p;
  ((float4*)r)[0] = q[0]; ((float4*)r)[1] = q[1];
  ((float4*)r)[2] = q[2]; ((float4*)r)[3] = q[3];
}

// ---------------------------------------------------------------- WMMA GEMM
// C[M,128] = A[M,128] @ W[128,128] + bias, optional fused skip-mix epilogue:
//   C = (A@W + b) * sigmoid(skip) + Xres * (1 - sigmoid(skip))
// Each 256-thread block stages the full 64KB W into LDS once via the Tensor
// Data Mover (TENSORcnt-tracked), then each wave pulls its 32 strip-invariant
// B fragments from LDS into registers ONCE, and sweeps 8 substrips of 16 rows
// with only A global loads + the 32-step V_WMMA_F32_16X16X4_F32 chain in the
// hot loop (no per-WMMA dscnt waits, deep load pipelining).
__global__ void gemm128_wmma_kernel(const float* __restrict__ A,
                                    const float* __restrict__ W,
                                    const float* __restrict__ bias,
                                    float* __restrict__ C,
                                    int M,
                                    const float* __restrict__ skip,
                                    const float* __restrict__ Xres) {
  __shared__ float Wlds[D * D];   // 64 KB (320 KB/WGP -> ~5 blocks resident)

#if defined(__gfx1250__)
  // ---- Tensor Data Mover: DMA the 128x128 f32 tensor into LDS (wave 0) ----
  if (threadIdx.x < 32) {
    unsigned long long ga = (unsigned long long)(const void*)W;
    unsigned lds_off = (unsigned)(size_t)(void*)&Wlds[0]; // generic LDS addr[31:0] = offset
    u32x4 g0;
    g0[0] = 1u;                                           // count=1, gather off
    g0[1] = lds_off;                                      // lds_addr (bytes)
    g0[2] = (unsigned)(ga & 0xFFFFFFFFu);                 // global_addr[31:0]
    g0[3] = ((unsigned)(ga >> 32) & 0x01FFFFFFu) | 0x80000000u; // addr[56:32] | type=2
    i32x8v g1;
    g1[0] = 0x00020000;        // data_size=4B (bits 17:16 = 2), no pad/iterate
    g1[1] = (int)(128u << 16); // tensor_dim0[15:0]=128 @ [63:48]
    g1[2] = (int)(128u << 16); // tensor_dim1[15:0]=128 @ [95:80]
    g1[3] = (int)(128u << 16); // tile_dim0=128 @ [127:112]
    g1[4] = 128;               // tile_dim1=128 @ [143:128]
    g1[5] = 128;               // tensor_dim0_stride[31:0]=128 @ [191:160]
    g1[6] = 0;
    g1[7] = 0;
    i32x4v z4; z4[0] = 0; z4[1] = 0; z4[2] = 0; z4[3] = 0;
#if __clang_major__ >= 23
    i32x8v z8; z8[0]=0; z8[1]=0; z8[2]=0; z8[3]=0; z8[4]=0; z8[5]=0; z8[6]=0; z8[7]=0;
    __builtin_amdgcn_tensor_load_to_lds(g0, g1, z4, z4, z8, 0);
#else
    __builtin_amdgcn_tensor_load_to_lds(g0, g1, z4, z4, 0);
#endif
    __builtin_amdgcn_s_wait_tensorcnt(0);
  }
#else
  for (int i = threadIdx.x; i < D * D / 4; i += 256)
    ((float4*)Wlds)[i] = ((const float4*)W)[i];
#endif
  __syncthreads();

  const int lane = threadIdx.x & 31;
  const int wave = threadIdx.x >> 5;     // 0..7 -> N tile
  const int n0   = wave * 16;
  const int r    = lane & 15;
  const int hi   = lane >> 4;            // 0: K pair {0,1}, 1: K pair {2,3}
  const int kSel = hi * 2;

  // Strip-invariant B fragments: read all 32 from LDS once, keep in VGPRs.
  // B fragment (4x16): lane<16 K=k0,k0+1 ; lane>=16 K=k0+2,k0+3 ; col n0+r.
  v2f bfrag[32];
#pragma unroll
  for (int kk = 0; kk < 32; ++kk) {
    const float* wp = &Wlds[(kk * 4 + kSel) * D + n0 + r];
    bfrag[kk][0] = wp[0];
    bfrag[kk][1] = wp[D];
  }

  float alpha = 1.f, beta = 0.f;
  if (skip) {
    float s = skip[0];
    alpha = 1.f / (1.f + __expf(-s));
    beta  = 1.f - alpha;
  }

  for (int sIdx = 0; sIdx < MSTRIPS; ++sIdx) {
    const int m0 = blockIdx.x * (16 * MSTRIPS) + sIdx * 16;
    if (m0 >= M) break;                                  // wave-uniform exit

    int rowA = m0 + r; if (rowA > M - 1) rowA = M - 1;   // clamp loads, EXEC stays full
    const float* Arow = A + (size_t)rowA * D;

    v8f acc = {};
#pragma unroll
    for (int kk = 0; kk < 32; ++kk) {
      // A fragment (16x4): lane<16 rows M=r K=4kk..4kk+1 ; lane>=16 K=4kk+2..4kk+3
      v2f af;
      af[0] = Arow[kk * 4 + kSel];
      af[1] = Arow[kk * 4 + kSel + 1];
      acc = __builtin_amdgcn_wmma_f32_16x16x4_f32(false, af, false, bfrag[kk],
                                                  (short)0, acc, false, false);
    }

    // C/D layout: lane<16 -> M=m0+i, N=n0+lane ; lane>=16 -> M=m0+8+i
    const int col   = n0 + r;
    const int rbase = m0 + hi * 8;
    const float bv  = bias[col];
#pragma unroll
    for (int i = 0; i < 8; ++i) {
      int row = rbase + i;
      if (row < M) {
        float v = acc[i] + bv;
        if (Xres) v = v * alpha + Xres[(size_t)row * D + col] * beta;
        C[(size_t)row * D + col] = v;
      }
    }
  }
}

// ---------------------------------------------------------------- edge phase
// pass 1: a[e,h] = (k[src] @ att[h]) . q[dst] * pri[h] / sqrt(DK); segment max
__global__ void attn_score_kernel(const float* __restrict__ ksrc,
                                  const float* __restrict__ qdst,
                                  const int* __restrict__ src,
                                  const int* __restrict__ dst,
                                  const float* __restrict__ att,
                                  const float* __restrict__ pri,
                                  float* __restrict__ abuf,
                                  float* __restrict__ amax,
                                  int E) {
  long long t = (long long)blockIdx.x * blockDim.x + threadIdx.x;
  if (t >= (long long)E * H) return;
  const int e = (int)(t >> 3), h = (int)(t & 7);
  const int s = src[e], d = dst[e];

  float kv[DK], qv[DK];
  load16(ksrc + (size_t)s * D + h * DK, kv);
  load16(qdst + (size_t)d * D + h * DK, qv);

  const float* ah = att + (size_t)h * DK * DK;
  float acc = 0.f;
#pragma unroll
  for (int dd = 0; dd < DK; ++dd) {
    float rv[DK];
    load16(ah + dd * DK, rv);
    float tv = 0.f;
#pragma unroll
    for (int f = 0; f < DK; ++f) tv += rv[f] * qv[f];
    acc += kv[dd] * tv;
  }
  const float aval = acc * pri[h] * 0.25f;   // 1/sqrt(16)
  abuf[t] = aval;
  atomicMaxF(&amax[(size_t)d * H + h], aval);
}

// pass 2: ea = exp(a - amax[dst]); segment sum of ea
__global__ void attn_norm_kernel(const int* __restrict__ dst,
                                 const float* __restrict__ amax,
                                 float* __restrict__ abuf,
                                 float* __restrict__ den,
                                 int E) {
  long long t = (long long)blockIdx.x * blockDim.x + threadIdx.x;
  if (t >= (long long)E * H) return;
  const int e = (int)(t >> 3), h = (int)(t & 7);
  const int d = dst[e];
  float m = amax[(size_t)d * H + h];
  if (!(m > -INFINITY)) m = 0.f;             // empty-segment guard (matches ref)
  const float ea = __expf(abuf[t] - m);
  abuf[t] = ea;
  atomicAdd(&den[(size_t)d * H + h], ea);
}

// pass 3: h[dst] += (ea/denom) * (v[src] @ msg[h])
__global__ void attn_agg_kernel(const float* __restrict__ vsrc,
                                const int* __restrict__ src,
                                const int* __restrict__ dst,
                                const float* __restrict__ msg,
                                const float* __restrict__ abuf,
                                const float* __restrict__ den,
                                float* __restrict__ hbuf,
                                int E) {
  long long t = (long long)blockIdx.x * blockDim.x + threadIdx.x;
  if (t >= (long long)E * H) return;
  const int e = (int)(t >> 3), h = (int)(t & 7);
  const int s = src[e], d = dst[e];
  const float w = abuf[t] / fmaxf(den[(size_t)d * H + h], 1e-9f);

  float vv[DK];
  load16(vsrc + (size_t)s * D + h * DK, vv);
  const float* mh = msg + (size_t)h * DK * DK;
  float* hout = hbuf + (size_t)d * D + h * DK;
#pragma unroll
  for (int f = 0; f < DK; ++f) {
    float acc = 0.f;
#pragma unroll
    for (int dd = 0; dd < DK; ++dd) acc += vv[dd] * mh[dd * DK + f];
    atomicAdd(&hout[f], w * acc);
  }
}

// t += scale * relu(h)
__global__ void relu_axpy_kernel(const float* __restrict__ h,
                                 float* __restrict__ t,
                                 float s, long long n) {
  long long i = (long long)blockIdx.x * blockDim.x + threadIdx.x;
  if (i < n) t[i] += s * fmaxf(h[i], 0.f);
}

// ---------------------------------------------------------------- launch
extern "C" void kernel_launch(void* const* d_in, const int* in_sizes, int n_in,
                              void* d_out, int out_size, void* d_ws, size_t ws_size,
                              hipStream_t stream) {
  const float* x_a = (const float*)d_in[0];
  const float* x_b = (const float*)d_in[1];
  const int* src_ab = (const int*)d_in[2];
  const int* dst_ab = (const int*)d_in[3];
  const int* src_ba = (const int*)d_in[4];
  const int* dst_ba = (const int*)d_in[5];
  const int* src_bb = (const int*)d_in[6];
  const int* dst_bb = (const int*)d_in[7];
  const float *Wk_a=(const float*)d_in[8],  *bk_a=(const float*)d_in[9];
  const float *Wq_a=(const float*)d_in[10], *bq_a=(const float*)d_in[11];
  const float *Wv_a=(const float*)d_in[12], *bv_a=(const float*)d_in[13];
  const float *Wa_a=(const float*)d_in[14], *ba_a=(const float*)d_in[15];
  const float *skip_a=(const float*)d_in[16];
  const float *Wk_b=(const float*)d_in[17], *bk_b=(const float*)d_in[18];
  const float *Wq_b=(const float*)d_in[19], *bq_b=(const float*)d_in[20];
  const float *Wv_b=(const float*)d_in[21], *bv_b=(const float*)d_in[22];
  const float *Wa_b=(const float*)d_in[23], *ba_b=(const float*)d_in[24];
  const float *skip_b=(const float*)d_in[25];
  const float *pri_ab=(const float*)d_in[26], *att_ab=(const float*)d_in[27], *msg_ab=(const float*)d_in[28];
  const float *pri_ba=(const float*)d_in[29], *att_ba=(const float*)d_in[30], *msg_ba=(const float*)d_in[31];
  const float *pri_bb=(const float*)d_in[32], *att_bb=(const float*)d_in[33], *msg_bb=(const float*)d_in[34];

  const int NA = in_sizes[0] / D;
  const int NB = in_sizes[1] / D;
  const int E  = in_sizes[2];
  const int NMAX = NA > NB ? NA : NB;

  // workspace carve-out (floats)
  float* ws = (float*)d_ws;
  size_t p = 0;
  auto take = [&](size_t n) { float* r = ws + p; p += n; return r; };
  float* ka = take((size_t)NA * D);
  float* qa = take((size_t)NA * D);
  float* va = take((size_t)NA * D);
  float* kb = take((size_t)NB * D);
  float* qb = take((size_t)NB * D);
  float* vb = take((size_t)NB * D);
  float* abuf = take((size_t)E * H);
  float* amax = take((size_t)NMAX * H);
  float* den  = take((size_t)NMAX * H);
  float* hbuf = take((size_t)NMAX * D);
  float* ta   = take((size_t)NA * D);
  float* tb   = take((size_t)NB * D);
  (void)ws_size; (void)n_in; (void)out_size;

  float* out_a = (float*)d_out;
  float* out_b = out_a + (size_t)NA * D;

  const dim3 blk(256);
  auto gb = [](long long n) { return dim3((unsigned)((n + 255) / 256)); };
  const int rowsPerBlk = 16 * MSTRIPS;
  const int gA = (NA + rowsPerBlk - 1) / rowsPerBlk;
  const int gB = (NB + rowsPerBlk - 1) / rowsPerBlk;

  // k/q/v projections (WMMA GEMMs, TDM-staged weights)
  gemm128_wmma_kernel<<<gA, blk, 0, stream>>>(x_a, Wk_a, bk_a, ka, NA, nullptr, nullptr);
  gemm128_wmma_kernel<<<gA, blk, 0, stream>>>(x_a, Wq_a, bq_a, qa, NA, nullptr, nullptr);
  gemm128_wmma_kernel<<<gA, blk, 0, stream>>>(x_a, Wv_a, bv_a, va, NA, nullptr, nullptr);
  gemm128_wmma_kernel<<<gB, blk, 0, stream>>>(x_b, Wk_b, bk_b, kb, NB, nullptr, nullptr);
  gemm128_wmma_kernel<<<gB, blk, 0, stream>>>(x_b, Wq_b, bq_b, qb, NB, nullptr, nullptr);
  gemm128_wmma_kernel<<<gB, blk, 0, stream>>>(x_b, Wv_b, bv_b, vb, NB, nullptr, nullptr);

  // zero the aggregation targets
  fill_kernel<<<gb((long long)NA * D), blk, 0, stream>>>(ta, 0.f, (long long)NA * D);
  fill_kernel<<<gb((long long)NB * D), blk, 0, stream>>>(tb, 0.f, (long long)NB * D);

  auto run_etype = [&](const int* src, const int* dst,
                       const float* ksrc, const float* qdst, const float* vsrc,
                       const float* att, const float* msg, const float* pri,
                       float* tacc, int ndst, float scale) {
    const long long nh = (long long)ndst * H;
    const long long nd = (long long)ndst * D;
    const long long eh = (long long)E * H;
    fill_kernel<<<gb(nh), blk, 0, stream>>>(amax, -INFINITY, nh);
    fill_kernel<<<gb(nh), blk, 0, stream>>>(den, 0.f, nh);
    fill_kernel<<<gb(nd), blk, 0, stream>>>(hbuf, 0.f, nd);
    attn_score_kernel<<<gb(eh), blk, 0, stream>>>(ksrc, qdst, src, dst, att, pri, abuf, amax, E);
    attn_norm_kernel<<<gb(eh), blk, 0, stream>>>(dst, amax, abuf, den, E);
    attn_agg_kernel<<<gb(eh), blk, 0, stream>>>(vsrc, src, dst, msg, abuf, den, hbuf, E);
    relu_axpy_kernel<<<gb(nd), blk, 0, stream>>>(hbuf, tacc, scale, nd);
  };

  // t_b = 0.5*(relu(agg_ab) + relu(agg_bb)) ; t_a = relu(agg_ba)
  run_etype(src_ab, dst_ab, ka, qb, va, att_ab, msg_ab, pri_ab, tb, NB, 0.5f);
  run_etype(src_bb, dst_bb, kb, qb, vb, att_bb, msg_bb, pri_bb, tb, NB, 0.5f);
  run_etype(src_ba, dst_ba, kb, qa, vb, att_ba, msg_ba, pri_ba, ta, NA, 1.0f);

  // output projections with fused sigmoid-skip epilogue
  gemm128_wmma_kernel<<<gA, blk, 0, stream>>>(ta, Wa_a, ba_a, out_a, NA, skip_a, x_a);
  gemm128_wmma_kernel<<<gB, blk, 0, stream>>>(tb, Wa_b, ba_b, out_b, NB, skip_b, x_b);
}